// CapsuleLayerWithRouting_23192823398469
// MI455X (gfx1250) — compile-verified
//
#include <hip/hip_runtime.h>
#include <math.h>

// ---------------------------------------------------------------------------
// CapsuleLayerWithRouting on MI455X (gfx1250)
//
// Strategy: never materialize U=[128,1152,32,16] (302MB). Recompute 16x16 U
// tiles with V_WMMA_F32_16X16X4_F32 in every pass; working set (x + W + B/C)
// ~85MB fits the 192MB L2, so routing passes stream from L2, not HBM.
// ---------------------------------------------------------------------------

typedef float v2f __attribute__((ext_vector_type(2)));
typedef float v8f __attribute__((ext_vector_type(8)));

#define BATCH 128
#define ICAPS 1152
#define IHID  16
#define OCAPS 32
#define OHID  16
#define ISPLIT 4
#define ICHUNK (ICAPS / ISPLIT)     // 288
#define WK (OCAPS * OHID)           // 512 floats: stride in W between k values

// ---------------------------------------------------------------------------
__global__ void zero_kernel(float* __restrict__ p, int n) {
    int i = blockIdx.x * blockDim.x + threadIdx.x;
    int stride = gridDim.x * blockDim.x;
    for (; i < n; i += stride) p[i] = 0.0f;
}

// C[b,i,:] = softmax(B[b,i,:]) over the 32 output capsules. One thread / row.
__global__ void softmax_kernel(const float* __restrict__ Bl,
                               float* __restrict__ Cl) {
    int row = blockIdx.x * blockDim.x + threadIdx.x;   // [0, BATCH*ICAPS)
    const float* br = Bl + (size_t)row * OCAPS;
    float*       cr = Cl + (size_t)row * OCAPS;
    float e[OCAPS];
    float m = -3.4e38f;
#pragma unroll
    for (int o = 0; o < OCAPS; ++o) { e[o] = br[o]; m = fmaxf(m, e[o]); }
    float s = 0.0f;
#pragma unroll
    for (int o = 0; o < OCAPS; ++o) { e[o] = __expf(e[o] - m); s += e[o]; }
    float inv = 1.0f / s;
#pragma unroll
    for (int o = 0; o < OCAPS; ++o) cr[o] = e[o] * inv;
}

// S[b,o,h] = sum_i C[b,i,o] * U[b,i,o,h],  U recomputed via WMMA.
// One wave owns (batch_tile of 16, one o, one i-chunk); c is folded into the
// A-fragment rows so the whole i-chunk accumulates in one 8-VGPR D register.
__global__ void __launch_bounds__(256)
s_kernel(const float* __restrict__ X, const float* __restrict__ W,
         const float* __restrict__ C, float* __restrict__ S) {
    int wid   = (blockIdx.x * blockDim.x + threadIdx.x) >> 5; // 0..ISPLIT*256-1
    int lane  = threadIdx.x & 31;
    int half  = lane >> 4;        // K-half select for A/B fragments
    int lm    = lane & 15;        // row (A) / column h (B)
    int o     = wid & (OCAPS - 1);
    int bt    = (wid >> 5) & 7;
    int split = wid >> 8;
    int b0    = bt * 16;
    int i0    = split * ICHUNK;

    v8f acc = {};
    const float* xrow  = X + ((size_t)(b0 + lm) * ICAPS + i0) * IHID;
    const float* cptr  = C + ((size_t)(b0 + lm) * ICAPS + i0) * OCAPS + o;
    const float* wbase = W + (size_t)i0 * IHID * WK + (size_t)o * OHID + lm;

    for (int i = 0; i < ICHUNK; ++i) {
        float c = *cptr;
        // A fragments: x[b0+lm, i, 4*kk + 2*half + {0,1}] (contiguous float2)
        v2f a0 = *(const v2f*)(xrow + 0  + 2 * half);
        v2f a1 = *(const v2f*)(xrow + 4  + 2 * half);
        v2f a2 = *(const v2f*)(xrow + 8  + 2 * half);
        v2f a3 = *(const v2f*)(xrow + 12 + 2 * half);
        // B fragments: W[i, k, o, lm], k = 4*kk + 2*half + {0,1}
        v2f w0, w1, w2, w3;
        w0.x = wbase[(0  + 2 * half) * WK];  w0.y = wbase[(1  + 2 * half) * WK];
        w1.x = wbase[(4  + 2 * half) * WK];  w1.y = wbase[(5  + 2 * half) * WK];
        w2.x = wbase[(8  + 2 * half) * WK];  w2.y = wbase[(9  + 2 * half) * WK];
        w3.x = wbase[(12 + 2 * half) * WK];  w3.y = wbase[(13 + 2 * half) * WK];
        // Fold routing coefficient into A rows: S = sum_i WMMA(c*x, W)
        v2f a0c = a0 * c, a1c = a1 * c, a2c = a2 * c, a3c = a3 * c;
        acc = __builtin_amdgcn_wmma_f32_16x16x4_f32(false, a0c, false, w0, (short)0, acc, false, false);
        acc = __builtin_amdgcn_wmma_f32_16x16x4_f32(false, a1c, false, w1, (short)0, acc, false, false);
        acc = __builtin_amdgcn_wmma_f32_16x16x4_f32(false, a2c, false, w2, (short)0, acc, false, false);
        acc = __builtin_amdgcn_wmma_f32_16x16x4_f32(false, a3c, false, w3, (short)0, acc, false, false);
        xrow += IHID; cptr += OCAPS; wbase += (size_t)IHID * WK;
    }
    // Merge i-splits into S (zeroed beforehand). D layout: VGPR v -> M=v+8*half, N=lm.
#pragma unroll
    for (int v = 0; v < 8; ++v) {
        int M = v + 8 * half;
        atomicAdd(&S[((size_t)(b0 + M) * OCAPS + o) * OHID + lm], acc[v]);
    }
}

// V[b,o,:] = squash(S[b,o,:]). One thread per (b,o) row.
__global__ void squash_kernel(const float* __restrict__ S,
                              float* __restrict__ V,
                              float* __restrict__ out, int write_out) {
    int row = blockIdx.x * blockDim.x + threadIdx.x;   // [0, BATCH*OCAPS)
    const float* s = S + (size_t)row * OHID;
    float e[OHID];
    float ns = 0.0f;
#pragma unroll
    for (int h = 0; h < OHID; ++h) { e[h] = s[h]; ns += e[h] * e[h]; }
    float scale = ns / ((1.0f + ns) * sqrtf(ns));
    float* vd = V + (size_t)row * OHID;
#pragma unroll
    for (int h = 0; h < OHID; ++h) {
        float val = e[h] * scale;
        vd[h] = val;
        if (write_out) out[(size_t)row * OHID + h] = val;
    }
}

// B[b,i,o] += sum_h U[b,i,o,h] * V[b,o,h], U recomputed via WMMA, then
// row-dot against preloaded V registers + half-wave shfl_xor reductions.
__global__ void __launch_bounds__(256)
bupd_kernel(const float* __restrict__ X, const float* __restrict__ W,
            const float* __restrict__ V, float* __restrict__ Bl) {
    int wid   = (blockIdx.x * blockDim.x + threadIdx.x) >> 5;
    int lane  = threadIdx.x & 31;
    int half  = lane >> 4;
    int lm    = lane & 15;
    int o     = wid & (OCAPS - 1);
    int bt    = (wid >> 5) & 7;
    int split = wid >> 8;
    int b0    = bt * 16;
    int i0    = split * ICHUNK;

    // Preload V tile in D-fragment layout (invariant over the i loop).
    float vv[8];
#pragma unroll
    for (int v = 0; v < 8; ++v) {
        int M = v + 8 * half;
        vv[v] = V[((size_t)(b0 + M) * OCAPS + o) * OHID + lm];
    }

    const float* xrow  = X + ((size_t)(b0 + lm) * ICAPS + i0) * IHID;
    const float* wbase = W + (size_t)i0 * IHID * WK + (size_t)o * OHID + lm;

    for (int i = 0; i < ICHUNK; ++i) {
        v8f acc = {};
        v2f a0 = *(const v2f*)(xrow + 0  + 2 * half);
        v2f a1 = *(const v2f*)(xrow + 4  + 2 * half);
        v2f a2 = *(const v2f*)(xrow + 8  + 2 * half);
        v2f a3 = *(const v2f*)(xrow + 12 + 2 * half);
        v2f w0, w1, w2, w3;
        w0.x = wbase[(0  + 2 * half) * WK];  w0.y = wbase[(1  + 2 * half) * WK];
        w1.x = wbase[(4  + 2 * half) * WK];  w1.y = wbase[(5  + 2 * half) * WK];
        w2.x = wbase[(8  + 2 * half) * WK];  w2.y = wbase[(9  + 2 * half) * WK];
        w3.x = wbase[(12 + 2 * half) * WK];  w3.y = wbase[(13 + 2 * half) * WK];
        acc = __builtin_amdgcn_wmma_f32_16x16x4_f32(false, a0, false, w0, (short)0, acc, false, false);
        acc = __builtin_amdgcn_wmma_f32_16x16x4_f32(false, a1, false, w1, (short)0, acc, false, false);
        acc = __builtin_amdgcn_wmma_f32_16x16x4_f32(false, a2, false, w2, (short)0, acc, false, false);
        acc = __builtin_amdgcn_wmma_f32_16x16x4_f32(false, a3, false, w3, (short)0, acc, false, false);

        // Row-wise dot: dB[M] = sum_N U[M,N]*V[M,N]; reduce N across each
        // 16-lane half (xor masks < 16 never cross halves).
        float mySum = 0.0f;
#pragma unroll
        for (int v = 0; v < 8; ++v) {
            float t = acc[v] * vv[v];
            t += __shfl_xor(t, 1, 32);
            t += __shfl_xor(t, 2, 32);
            t += __shfl_xor(t, 4, 32);
            t += __shfl_xor(t, 8, 32);
            if (lm == v) mySum = t;   // lane lm holds row (8*half + lm)
        }
        if (lm < 8) {
            int rowb = 8 * half + lm;
            size_t idx = ((size_t)(b0 + rowb) * ICAPS + (size_t)(i0 + i)) * OCAPS + o;
            Bl[idx] += mySum;         // exclusive owner: no atomics needed
        }
        xrow += IHID; wbase += (size_t)IHID * WK;
    }
}

// ---------------------------------------------------------------------------
extern "C" void kernel_launch(void* const* d_in, const int* in_sizes, int n_in,
                              void* d_out, int out_size, void* d_ws, size_t ws_size,
                              hipStream_t stream) {
    const float* X = (const float*)d_in[0];   // [128,1152,16]
    const float* W = (const float*)d_in[1];   // [1152,16,32,16]
    // d_in[2] = routing_iterations (fixed at 3 by setup; structure hardcoded)
    float* out = (float*)d_out;               // V: [128,32,16]

    float* Bbuf = (float*)d_ws;                       // 128*1152*32
    size_t nB = (size_t)BATCH * ICAPS * OCAPS;
    float* Cbuf = Bbuf + nB;                          // 128*1152*32
    float* Sbuf = Cbuf + nB;                          // 128*32*16
    float* Vbuf = Sbuf + (size_t)BATCH * OCAPS * OHID;

    zero_kernel<<<2048, 256, 0, stream>>>(Bbuf, (int)nB);
    for (int it = 0; it < 3; ++it) {
        softmax_kernel<<<(BATCH * ICAPS) / 256, 256, 0, stream>>>(Bbuf, Cbuf);
        zero_kernel<<<64, 256, 0, stream>>>(Sbuf, BATCH * OCAPS * OHID);
        s_kernel<<<ISPLIT * 32, 256, 0, stream>>>(X, W, Cbuf, Sbuf);
        squash_kernel<<<(BATCH * OCAPS) / 256, 256, 0, stream>>>(Sbuf, Vbuf, out, it == 2);
        if (it < 2)
            bupd_kernel<<<ISPLIT * 32, 256, 0, stream>>>(X, W, Vbuf, Bbuf);
    }
}